// GraphLSTM_1477468750569
// MI455X (gfx1250) — compile-verified
//
#include <hip/hip_runtime.h>
#include <hip/hip_bf16.h>

// ---------------------------------------------------------------------------
// GraphLSTM (GCN x2 -> LSTMCell scan -> big Linear) for MI455X / gfx1250.
//
// Roofline reasoning:
//  * Final linear reads W_lin = 256 MB f32 -> memory bound (~11us @ 23.3TB/s).
//    Kept in f32: per-call bf16 conversion would add traffic, not remove it.
//  * Per-step GEMMs are tiny & on the serial scan path -> latency bound; all
//    mapped to wave-level 16x16 tiles via V_WMMA_F32_16X16X4_F32 (wave32).
//  * OOB handling: A-row / B-col indices are CLAMPED (not masked). A garbage
//    row m only affects D row m, which the guarded store never writes, so the
//    inner loop has zero branches / exec-mask toggles -> plain b64 loads.
//  * GCN aggregation = float atomics; self-loops folded into GEMM epilogue.
// ---------------------------------------------------------------------------

typedef __attribute__((ext_vector_type(2))) float v2f;
typedef __attribute__((ext_vector_type(8))) float v8f;

#define TDIM 48
#define NNODE 1000
#define DIN 32
#define HDIM 64
#define GDIM (4 * HDIM)   // 256

__device__ __forceinline__ v8f wmma_f32_16x16x4(v2f a, v2f b, v8f c) {
  // 8 args: (neg_a, A, neg_b, B, c_mod, C, reuse_a, reuse_b)
  return __builtin_amdgcn_wmma_f32_16x16x4_f32(false, a, false, b, (short)0, c,
                                               false, false);
}

// ---------------------------------------------------------------------------
// C[M x N] = A[M x K] @ B[K x N]   (B row-major, ldb = N stride)
// Optionally writes Cself[m,n] = C[m,n] * dinv[m]^2  (GCN self-loop term).
// One wave computes one 16x16 tile with f32 WMMA (K stepped by 4).
// ---------------------------------------------------------------------------
__global__ void wmma_gemm_nn(const float* __restrict__ A, int lda,
                             const float* __restrict__ B, int ldb,
                             float* __restrict__ C,
                             float* __restrict__ Cself,
                             const float* __restrict__ dinv,
                             int M, int N, int K) {
  const int tilesN = (N + 15) >> 4;
  const int tilesM = (M + 15) >> 4;
  const int wid = blockIdx.x * (blockDim.x >> 5) + (threadIdx.x >> 5);
  if (wid >= tilesM * tilesN) return;
  const int mt = wid / tilesN;
  const int nt = wid - mt * tilesN;
  const int m0 = mt << 4, n0 = nt << 4;
  const int lane = threadIdx.x & 31;
  const int lm = lane & 15;           // A row / B col within tile
  const int kk = (lane >> 4) << 1;    // lanes 0-15 -> K0,K1 ; lanes 16-31 -> K2,K3
  const int ar = m0 + lm;
  const int bc = n0 + lm;
  // clamp into bounds; garbage rows/cols never stored (see header comment)
  const int arc = (ar < M) ? ar : (M - 1);
  const int bcc = (bc < N) ? bc : (N - 1);

  const float* ap = A + (size_t)arc * lda + kk;   // += 4 per step
  const float* bp = B + (size_t)kk * ldb + bcc;   // += 4*ldb per step
  const size_t bstep = (size_t)4 * ldb;

  v8f acc = {};
#pragma unroll 4
  for (int k0 = 0; k0 < K; k0 += 4) {
    v2f a = *(const v2f*)ap;
    v2f b;
    b.x = bp[0];
    b.y = bp[ldb];
    acc = wmma_f32_16x16x4(a, b, acc);
    ap += 4;
    bp += bstep;
  }
  // C/D layout: VGPR r -> row m0+r (lanes 0-15) / m0+8+r (lanes 16-31)
  const int col = n0 + lm;
  const int rbase = m0 + ((lane >> 4) << 3);
#pragma unroll
  for (int r = 0; r < 8; ++r) {
    const int row = rbase + r;
    if (row < M && col < N) {
      const float v = acc[r];
      C[(size_t)row * N + col] = v;
      if (Cself) {
        const float dv = dinv[row];
        Cself[(size_t)row * N + col] = v * dv * dv;
      }
    }
  }
}

// ---------------------------------------------------------------------------
// C[M x N] += A[M x K] @ B^T  where B is [N x K] row-major (ldb = K stride).
// Split-K over blockIdx.y (chunkK elements each); results atomicAdd'ed so C
// must be pre-initialized (with bias). Used for LSTM gates + final linear.
// ---------------------------------------------------------------------------
__global__ void wmma_gemm_nt_atomic(const float* __restrict__ A, int lda,
                                    const float* __restrict__ B, int ldb,
                                    float* __restrict__ C, int ldc,
                                    int M, int N, int K, int chunkK) {
  const int tilesN = (N + 15) >> 4;
  const int tilesM = (M + 15) >> 4;
  const int wid = blockIdx.x * (blockDim.x >> 5) + (threadIdx.x >> 5);
  if (wid >= tilesM * tilesN) return;
  const int mt = wid / tilesN;
  const int nt = wid - mt * tilesN;
  const int m0 = mt << 4, n0 = nt << 4;
  const int lane = threadIdx.x & 31;
  const int lm = lane & 15;
  const int kk = (lane >> 4) << 1;
  const int ar = m0 + lm;
  const int bc = n0 + lm;
  const int arc = (ar < M) ? ar : (M - 1);
  const int bcc = (bc < N) ? bc : (N - 1);

  const int kbeg = blockIdx.y * chunkK;
  int kend = kbeg + chunkK;
  if (kend > K) kend = K;

  const float* ap = A + (size_t)arc * lda + kbeg + kk;  // += 4 per step
  const float* bp = B + (size_t)bcc * ldb + kbeg + kk;  // += 4 per step

  v8f acc = {};
#pragma unroll 4
  for (int k0 = kbeg; k0 < kend; k0 += 4) {
    v2f a = *(const v2f*)ap;
    v2f b = *(const v2f*)bp;
    acc = wmma_f32_16x16x4(a, b, acc);
    ap += 4;
    bp += 4;
  }
  const int col = n0 + lm;
  const int rbase = m0 + ((lane >> 4) << 3);
#pragma unroll
  for (int r = 0; r < 8; ++r) {
    const int row = rbase + r;
    if (row < M && col < N) atomicAdd(&C[(size_t)row * ldc + col], acc[r]);
  }
}

// --------------------------- GCN normalization ------------------------------
__global__ void deg_init(float* __restrict__ deg, int n) {
  const int i = blockIdx.x * blockDim.x + threadIdx.x;
  if (i < n) deg[i] = 1.0f;  // self loop
}
__global__ void deg_accum(float* __restrict__ deg, const int* __restrict__ ei,
                          int E) {
  const int e = blockIdx.x * blockDim.x + threadIdx.x;
  if (e < E) atomicAdd(&deg[ei[E + e]], 1.0f);  // dst = ei[1][e]
}
__global__ void deg_inv(const float* __restrict__ deg, float* __restrict__ dinv,
                        int n) {
  const int i = blockIdx.x * blockDim.x + threadIdx.x;
  if (i < n) dinv[i] = rsqrtf(deg[i]);  // deg >= 1 guaranteed
}

// msg = feat[src] * dinv[src]*dinv[dst], scattered into acc[dst]
__global__ void edge_scatter(const float* __restrict__ src_feat,
                             float* __restrict__ acc,
                             const int* __restrict__ ei,
                             const float* __restrict__ dinv, int E) {
  const int idx = blockIdx.x * blockDim.x + threadIdx.x;
  if (idx >= E * HDIM) return;
  const int e = idx >> 6;   // / HDIM
  const int f = idx & 63;   // % HDIM
  const int s = ei[e];
  const int d = ei[E + e];
  const float nrm = dinv[s] * dinv[d];
  atomicAdd(&acc[d * HDIM + f], src_feat[s * HDIM + f] * nrm);
}

__global__ void gcn_finalize(const float* __restrict__ acc,
                             const float* __restrict__ bias,
                             float* __restrict__ out, int total, int relu) {
  const int i = blockIdx.x * blockDim.x + threadIdx.x;
  if (i >= total) return;
  float v = acc[i] + bias[i & 63];
  out[i] = relu ? fmaxf(v, 0.0f) : v;
}

// --------------------------------- LSTM -------------------------------------
__global__ void gates_init(float* __restrict__ gates,
                           const float* __restrict__ b_ih,
                           const float* __restrict__ b_hh, int total) {
  const int i = blockIdx.x * blockDim.x + threadIdx.x;
  if (i >= total) return;
  const int j = i & (GDIM - 1);
  gates[i] = b_ih[j] + b_hh[j];
}

__device__ __forceinline__ float sigf(float x) {
  return 1.0f / (1.0f + __expf(-x));
}

__global__ void lstm_cell(const float* __restrict__ gates,
                          float* __restrict__ c, float* __restrict__ h,
                          float* __restrict__ hseq_t, int NH) {
  const int i = blockIdx.x * blockDim.x + threadIdx.x;
  if (i >= NH) return;
  const int n = i >> 6;
  const int j = i & 63;
  const float* gr = gates + n * GDIM;
  const float gi = gr[j];
  const float gf = gr[HDIM + j];
  const float gg = gr[2 * HDIM + j];
  const float go = gr[3 * HDIM + j];
  const float cn = sigf(gf) * c[i] + sigf(gi) * tanhf(gg);
  const float hn = sigf(go) * tanhf(cn);
  c[i] = cn;
  h[i] = hn;
  hseq_t[i] = hn;
}

// ------------------------------- misc ---------------------------------------
__global__ void out_init(float* __restrict__ out,
                         const float* __restrict__ b_lin, int total, int n) {
  const int i = blockIdx.x * blockDim.x + threadIdx.x;
  if (i < total) out[i] = b_lin[i % n];
}
__global__ void vec_copy(float* __restrict__ dst, const float* __restrict__ src,
                         int n) {
  const int i = blockIdx.x * blockDim.x + threadIdx.x;
  if (i < n) dst[i] = src[i];
}

// ---------------------------------------------------------------------------
extern "C" void kernel_launch(void* const* d_in, const int* in_sizes, int n_in,
                              void* d_out, int out_size, void* d_ws,
                              size_t ws_size, hipStream_t stream) {
  const float* x    = (const float*)d_in[0];   // [T, N, DIN]
  const int* ei     = (const int*)d_in[1];     // [2, E]
  const float* h0   = (const float*)d_in[2];   // [N, H]
  const float* c0   = (const float*)d_in[3];   // [N, H]
  const float* Wg1  = (const float*)d_in[4];   // [DIN, H]
  const float* bg1  = (const float*)d_in[5];   // [H]
  const float* Wg2  = (const float*)d_in[6];   // [H, H]
  const float* bg2  = (const float*)d_in[7];   // [H]
  const float* W_ih = (const float*)d_in[8];   // [4H, H]
  const float* W_hh = (const float*)d_in[9];   // [4H, H]
  const float* b_ih = (const float*)d_in[10];  // [4H]
  const float* b_hh = (const float*)d_in[11];  // [4H]
  const float* Wlin = (const float*)d_in[12];  // [N, H*N]
  const float* blin = (const float*)d_in[13];  // [N]
  float* out = (float*)d_out;                  // [T*N] ++ h[N*H] ++ c[N*H]

  const int E = in_sizes[1] / 2;
  const int NH = NNODE * HDIM;        // 64000
  const int KLIN = NH;                // final GEMM K

  // Workspace layout (floats); total ~3.72M floats (~15 MB).
  float* ws    = (float*)d_ws;
  float* deg   = ws;                  // 1000
  float* dinv  = ws + 1000;           // 1000
  float* tmp1  = ws + 2000;           // 64000  (GEMM out, per-src features)
  float* tmp2  = ws + 66000;          // 64000  (scatter accumulator)
  float* g1    = ws + 130000;         // 64000  (relu conv1 out)
  float* g2    = ws + 194000;         // 64000  (conv2 out)
  float* h     = ws + 258000;         // 64000
  float* c     = ws + 322000;         // 64000
  float* gates = ws + 386000;         // 256000
  float* hseq  = ws + 642000;         // T*64000 = 3072000

  const int B256 = 256;
  auto cdiv = [](int a, int b) { return (a + b - 1) / b; };
  // GEMM launch helper: 4 waves (128 thr) per block, 1 tile per wave.
  auto gemm_blocks = [&](int M, int N) {
    return cdiv(cdiv(M, 16) * cdiv(N, 16), 4);
  };

  // ---- degree / symmetric norm (once per call) ----
  deg_init<<<cdiv(NNODE, B256), B256, 0, stream>>>(deg, NNODE);
  deg_accum<<<cdiv(E, B256), B256, 0, stream>>>(deg, ei, E);
  deg_inv<<<cdiv(NNODE, B256), B256, 0, stream>>>(deg, dinv, NNODE);

  // ---- initial hidden / cell ----
  vec_copy<<<cdiv(NH, B256), B256, 0, stream>>>(h, h0, NH);
  vec_copy<<<cdiv(NH, B256), B256, 0, stream>>>(c, c0, NH);

  // ---- time scan ----
  for (int t = 0; t < TDIM; ++t) {
    const float* xt = x + (size_t)t * NNODE * DIN;

    // GCN conv1: tmp1 = xt @ Wg1 ; tmp2 = tmp1 * dinv^2 (self loop)
    wmma_gemm_nn<<<gemm_blocks(NNODE, HDIM), 128, 0, stream>>>(
        xt, DIN, Wg1, HDIM, tmp1, tmp2, dinv, NNODE, HDIM, DIN);
    edge_scatter<<<cdiv(E * HDIM, B256), B256, 0, stream>>>(tmp1, tmp2, ei,
                                                            dinv, E);
    gcn_finalize<<<cdiv(NH, B256), B256, 0, stream>>>(tmp2, bg1, g1, NH, 1);

    // GCN conv2
    wmma_gemm_nn<<<gemm_blocks(NNODE, HDIM), 128, 0, stream>>>(
        g1, HDIM, Wg2, HDIM, tmp1, tmp2, dinv, NNODE, HDIM, HDIM);
    edge_scatter<<<cdiv(E * HDIM, B256), B256, 0, stream>>>(tmp1, tmp2, ei,
                                                            dinv, E);
    gcn_finalize<<<cdiv(NH, B256), B256, 0, stream>>>(tmp2, bg2, g2, NH, 0);

    // LSTM gates = g2 @ W_ih^T + h @ W_hh^T + b_ih + b_hh
    gates_init<<<cdiv(NNODE * GDIM, B256), B256, 0, stream>>>(gates, b_ih,
                                                              b_hh,
                                                              NNODE * GDIM);
    {
      dim3 grid(gemm_blocks(NNODE, GDIM), 1);
      wmma_gemm_nt_atomic<<<grid, 128, 0, stream>>>(
          g2, HDIM, W_ih, HDIM, gates, GDIM, NNODE, GDIM, HDIM, HDIM);
      wmma_gemm_nt_atomic<<<grid, 128, 0, stream>>>(
          h, HDIM, W_hh, HDIM, gates, GDIM, NNODE, GDIM, HDIM, HDIM);
    }
    lstm_cell<<<cdiv(NH, B256), B256, 0, stream>>>(
        gates, c, h, hseq + (size_t)t * NH, NH);
  }

  // ---- final linear: out[t,n] = hseq[t,:] . Wlin[n,:] + blin[n] ----
  out_init<<<cdiv(TDIM * NNODE, B256), B256, 0, stream>>>(out, blin,
                                                          TDIM * NNODE, NNODE);
  {
    const int KSPLIT = 64;                 // 64 chunks of 1000 (multiple of 4)
    dim3 grid(gemm_blocks(TDIM, NNODE), KSPLIT);
    wmma_gemm_nt_atomic<<<grid, 128, 0, stream>>>(
        hseq, KLIN, Wlin, KLIN, out, NNODE, TDIM, NNODE, KLIN, KLIN / KSPLIT);
  }

  // ---- trailing outputs: final hidden & cell state ----
  vec_copy<<<cdiv(NH, B256), B256, 0, stream>>>(out + TDIM * NNODE, h, NH);
  vec_copy<<<cdiv(NH, B256), B256, 0, stream>>>(out + TDIM * NNODE + NH, c, NH);
}